// Net_GCN_Eigen_GlobalPooling_56727928046285
// MI455X (gfx1250) — compile-verified
//
#include <hip/hip_runtime.h>
#include <hip/hip_bf16.h>

typedef __attribute__((ext_vector_type(2))) float v2f;
typedef __attribute__((ext_vector_type(8))) float v8f;

#define EPS 1e-5f

// ---------------------------------------------------------------------------
// Pad x[:, :39] (stride 42) into Xpad (stride 40), col 39 = 0
// ---------------------------------------------------------------------------
__global__ void pad_x_kernel(const float* __restrict__ x, float* __restrict__ xp, int n) {
    int total = n * 40;
    for (int i = blockIdx.x * blockDim.x + threadIdx.x; i < total; i += gridDim.x * blockDim.x) {
        int r = i / 40, c = i - r * 40;
        xp[i] = (c < 39) ? x[(size_t)r * 42 + c] : 0.0f;
    }
}

// ---------------------------------------------------------------------------
// WMMA fp32 GEMM:  Y[n x fo] = A[n x kpad] * B[kact x fo]
// Block = 128 threads (4 waves). Block tile = 128 rows x 64 cols.
// Wave w -> rows [row0+32w, row0+32w+32) as two 16-row M tiles, all 64 cols.
// B stripe staged in LDS, pair-interleaved: sBp[p*64+n] = {B[2p][n], B[2p+1][n]}
// so a B fragment is one aligned ds_load_b64 into an adjacent VGPR pair, and
// each fragment feeds two WMMAs (both M tiles).
// A 16x4 f32 frag: a[g] = A[m][k + 2*hi + g], m = lane&15, hi = lane>>4
// B 4x16  frag:    b[g] = B[k + 2*hi + g][lane&15]  (pair index k/2 + hi)
// D 16x16 frag:    d[g] at row (g + 8*hi), col (lane&15)
// ---------------------------------------------------------------------------
__global__ __launch_bounds__(128) void gemm_wmma_f32(
    const float* __restrict__ A, int lda,
    const float* __restrict__ B, int ldb, int kact,
    float* __restrict__ Y, int ldy, int kpad)
{
    __shared__ float2 sBp[128 * 64];   // (kpad/2) x 64 pairs, <= 64 KB

    const int row0 = blockIdx.x * 128;
    const int col0 = blockIdx.y * 64;
    const int tid  = threadIdx.x;

    // cooperative load of B stripe into LDS, pair-interleaved (zero-fill pad)
    const int npair = (kpad >> 1) * 64;
    for (int i = tid; i < npair; i += 128) {
        int p = i >> 6, c = i & 63;
        int k0 = 2 * p;
        float2 v;
        v.x = (k0     < kact) ? B[(size_t)k0 * ldb + col0 + c]       : 0.0f;
        v.y = (k0 + 1 < kact) ? B[(size_t)(k0 + 1) * ldb + col0 + c] : 0.0f;
        sBp[i] = v;
    }
    __syncthreads();

    const int wave = tid >> 5;
    const int lane = tid & 31;
    const int hi   = lane >> 4;
    const int lm   = lane & 15;

    const float* __restrict__ Arow0 = A + (size_t)(row0 + wave * 32 + lm) * lda;
    const float* __restrict__ Arow1 = Arow0 + (size_t)16 * lda;
    const float2* __restrict__ sbase = sBp + hi * 64 + lm;

    v8f acc[2][4];
    const v8f vz = {0.f, 0.f, 0.f, 0.f, 0.f, 0.f, 0.f, 0.f};
#pragma unroll
    for (int mt = 0; mt < 2; ++mt)
#pragma unroll
        for (int nt = 0; nt < 4; ++nt) acc[mt][nt] = vz;

    for (int k = 0; k < kpad; k += 4) {
        // A fragments: two consecutive floats, 8B aligned (lda even, k%4==0)
        const float2 av0 = *reinterpret_cast<const float2*>(Arow0 + k + 2 * hi);
        const float2 av1 = *reinterpret_cast<const float2*>(Arow1 + k + 2 * hi);
        v2f a0; a0.x = av0.x; a0.y = av0.y;
        v2f a1; a1.x = av1.x; a1.y = av1.y;
        const float2* srow = sbase + (k >> 1) * 64;
#pragma unroll
        for (int nt = 0; nt < 4; ++nt) {
            const float2 bv = srow[nt * 16];     // one ds_load_b64
            v2f b; b.x = bv.x; b.y = bv.y;
            acc[0][nt] = __builtin_amdgcn_wmma_f32_16x16x4_f32(
                false, a0, false, b, (short)0, acc[0][nt], false, false);
            acc[1][nt] = __builtin_amdgcn_wmma_f32_16x16x4_f32(
                false, a1, false, b, (short)0, acc[1][nt], false, false);
        }
    }

#pragma unroll
    for (int mt = 0; mt < 2; ++mt) {
#pragma unroll
        for (int nt = 0; nt < 4; ++nt) {
#pragma unroll
            for (int g = 0; g < 8; ++g) {
                Y[(size_t)(row0 + wave * 32 + mt * 16 + g + 8 * hi) * ldy
                  + col0 + nt * 16 + lm] = acc[mt][nt][g];
            }
        }
    }
}

// ---------------------------------------------------------------------------
// H[i] = relu(Y[prev(i)] + Y[next(i)] + bias), ring wrap within segment S
// (S is a power of two). Per-block column partial sums/sumsq written to
// partials[block*2F + c] / [.. + F + c] for a deterministic reduction.
// Block = F threads, rows_per_block = 256.
// ---------------------------------------------------------------------------
__global__ void neigh_relu_stats(
    const float* __restrict__ Y, const float* __restrict__ bias,
    float* __restrict__ H, float* __restrict__ partials,
    int F, int S, int rowsPerBlock)
{
    const int c    = threadIdx.x;
    const int r0   = blockIdx.x * rowsPerBlock;
    const int mask = S - 1;
    const float bv = bias[c];
    float s = 0.0f, s2 = 0.0f;
    for (int r = r0; r < r0 + rowsPerBlock; ++r) {
        int loc  = r & mask;
        int base = r - loc;
        int prev = base + ((loc + S - 1) & mask);
        int next = base + ((loc + 1) & mask);
        float h = Y[(size_t)prev * F + c] + Y[(size_t)next * F + c] + bv;
        h = fmaxf(h, 0.0f);
        H[(size_t)r * F + c] = h;
        s += h; s2 += h * h;
    }
    partials[(size_t)blockIdx.x * 2 * F + c]     = s;
    partials[(size_t)blockIdx.x * 2 * F + F + c] = s2;
}

// Deterministic fixed-order reduction of partials -> stats[c]=sum, stats[256+c]=sumsq
__global__ void reduce_stats(const float* __restrict__ partials,
                             float* __restrict__ stats, int nb, int F)
{
    const int c = threadIdx.x;
    float s = 0.0f, s2 = 0.0f;
    for (int b = 0; b < nb; ++b) {
        s  += partials[(size_t)b * 2 * F + c];
        s2 += partials[(size_t)b * 2 * F + F + c];
    }
    stats[c]       = s;
    stats[256 + c] = s2;
}

// BatchNorm (training stats) applied in place.  Block = F threads.
__global__ void bn_apply(float* __restrict__ H,
                         const float* __restrict__ g, const float* __restrict__ beta,
                         const float* __restrict__ stats,
                         int n, int F, int rowsPerBlock)
{
    const int c = threadIdx.x;
    const float invn  = 1.0f / (float)n;
    const float mean  = stats[c] * invn;
    const float var   = stats[256 + c] * invn - mean * mean;
    const float scale = g[c] * rsqrtf(var + EPS);
    const float shift = beta[c] - mean * scale;
    const int r0 = blockIdx.x * rowsPerBlock;
    for (int r = r0; r < r0 + rowsPerBlock; ++r) {
        size_t idx = (size_t)r * F + c;
        H[idx] = H[idx] * scale + shift;
    }
}

// BatchNorm + max-pool over groups of 4 consecutive rows.  Block = F threads.
__global__ void bn_pool(const float* __restrict__ H,
                        const float* __restrict__ g, const float* __restrict__ beta,
                        const float* __restrict__ stats,
                        float* __restrict__ Xout,
                        int n, int F, int outRowsPerBlock)
{
    const int c = threadIdx.x;
    const float invn  = 1.0f / (float)n;
    const float mean  = stats[c] * invn;
    const float var   = stats[256 + c] * invn - mean * mean;
    const float scale = g[c] * rsqrtf(var + EPS);
    const float shift = beta[c] - mean * scale;
    const int j0 = blockIdx.x * outRowsPerBlock;
    for (int j = j0; j < j0 + outRowsPerBlock; ++j) {
        float m = -3.4e38f;
#pragma unroll
        for (int q = 0; q < 4; ++q) {
            float v = H[(size_t)(4 * j + q) * F + c] * scale + shift;
            m = fmaxf(m, v);
        }
        Xout[(size_t)j * F + c] = m;
    }
}

// ---------------------------------------------------------------------------
// Head: per-graph mean (512 rows of 256 feats) -> relu(xg@W1+b1)@W2+b2
// Grid = 8 blocks (one per graph), 256 threads.
// ---------------------------------------------------------------------------
__global__ void fc_head(const float* __restrict__ X3,
                        const float* __restrict__ W1, const float* __restrict__ b1,
                        const float* __restrict__ W2, const float* __restrict__ b2,
                        float* __restrict__ out)
{
    __shared__ float xg[256];
    __shared__ float h1[256];
    const int g = blockIdx.x;
    const int c = threadIdx.x;

    float s = 0.0f;
    for (int r = 0; r < 512; ++r) s += X3[(size_t)(g * 512 + r) * 256 + c];
    xg[c] = s * (1.0f / 512.0f);
    __syncthreads();

    float a = b1[c];
    for (int k = 0; k < 256; ++k) a += xg[k] * W1[(size_t)k * 256 + c];
    h1[c] = fmaxf(a, 0.0f);
    __syncthreads();

    if (c < 30) {
        float o = b2[c];
        for (int j = 0; j < 256; ++j) o += h1[j] * W2[(size_t)j * 30 + c];
        out[(size_t)g * 30 + c] = o;
    }
}

// ---------------------------------------------------------------------------
extern "C" void kernel_launch(void* const* d_in, const int* in_sizes, int n_in,
                              void* d_out, int out_size, void* d_ws, size_t ws_size,
                              hipStream_t stream) {
    (void)n_in; (void)out_size; (void)ws_size;
    const float* x = (const float*)d_in[0];

    // Resolve parameter pointers. Pytree flattening may order the params dict
    // 'convs','bns','fc' (insertion order) or 'bns','convs','fc' (sorted keys).
    // Discriminate via in_sizes[3]: first conv W = 39*64 = 2496, first bn g = 64.
    const float *convW[12], *convB[12], *bnG[12], *bnB[12];
    int idx = 3;
    bool convs_first = (in_sizes[3] == 39 * 64);
    if (convs_first) {
        for (int i = 0; i < 12; ++i) { convW[i] = (const float*)d_in[idx++]; convB[i] = (const float*)d_in[idx++]; }
        for (int i = 0; i < 12; ++i) { bnG[i]   = (const float*)d_in[idx++]; bnB[i]   = (const float*)d_in[idx++]; }
    } else {
        for (int i = 0; i < 12; ++i) { bnG[i]   = (const float*)d_in[idx++]; bnB[i]   = (const float*)d_in[idx++]; }
        for (int i = 0; i < 12; ++i) { convW[i] = (const float*)d_in[idx++]; convB[i] = (const float*)d_in[idx++]; }
    }
    const float* fcW1 = (const float*)d_in[idx++];
    const float* fcB1 = (const float*)d_in[idx++];
    const float* fcW2 = (const float*)d_in[idx++];
    const float* fcB2 = (const float*)d_in[idx++];

    // Workspace layout
    char* ws = (char*)d_ws;
    float* stats    = (float*)ws;                                  // 12 convs x 512 floats
    float* partials = (float*)(ws + 32 * 1024);                    // up to 1024 blocks x 512 floats = 2 MB
    size_t slabBytes = (size_t)262144 * 128 * sizeof(float);       // 134 MB
    float* slabA = (float*)(ws + 32 * 1024 + 2 * 1024 * 1024);
    float* slabB = (float*)(ws + 32 * 1024 + 2 * 1024 * 1024 + slabBytes);

    // Level / conv configuration (K padded to multiple of 4 for WMMA)
    const int levN[4]  = {262144, 65536, 16384, 4096};   // nodes per level
    const int levS[4]  = {32768, 8192, 2048, 512};       // ring period (per graph)
    const int FinP[12] = {40, 64, 64, 128, 128, 128, 128, 256, 256, 256, 256, 256};
    const int Kact[12] = {39, 64, 64, 128, 128, 128, 128, 256, 256, 256, 256, 256};
    const int Fout[12] = {64, 64, 128, 128, 128, 128, 256, 256, 256, 256, 256, 256};

    // Stage padded level-0 input into slabA
    pad_x_kernel<<<1024, 256, 0, stream>>>(x, slabA, 262144);

    float* X = slabA;   // current level input
    float* O = slabB;   // GEMM output / scratch

    for (int L = 0; L < 4; ++L) {
        const int n = levN[L], S = levS[L];
        for (int c3 = 0; c3 < 3; ++c3) {
            const int j   = L * 3 + c3;
            const int fin = FinP[j], ka = Kact[j], fo = Fout[j];

            // Y = X @ W   (X in slab holding X; Y into the other slab)
            dim3 grid(n / 128, fo / 64);
            gemm_wmma_f32<<<grid, 128, 0, stream>>>(X, fin, convW[j], fo, ka, O, fo, fin);

            // H = relu(shift(Y)+b) written over the X slab (X is dead after GEMM)
            const int rpb = 256;
            const int nb  = n / rpb;
            neigh_relu_stats<<<nb, fo, 0, stream>>>(O, convB[j], X, partials, fo, S, rpb);
            reduce_stats<<<1, fo, 0, stream>>>(partials, stats + j * 512, nb, fo);

            if (c3 == 2 && L < 3) {
                // BN + pool(4): reads H (X slab), writes pooled level input into O slab
                const int orpb = 64;
                bn_pool<<<(n / 4) / orpb, fo, 0, stream>>>(X, bnG[j], bnB[j], stats + j * 512,
                                                           O, n, fo, orpb);
                float* t = X; X = O; O = t;   // pooled X now lives in the other slab
            } else {
                // BN in place on H (X slab): next conv reads it as X
                bn_apply<<<nb, fo, 0, stream>>>(X, bnG[j], bnB[j], stats + j * 512, n, fo, rpb);
            }
        }
    }

    // Per-graph mean + 2-layer MLP head -> (8, 30)
    fc_head<<<8, 256, 0, stream>>>(X, fcW1, fcB1, fcW2, fcB2, (float*)d_out);
}